// ROIAlignLayer_41042707480973
// MI455X (gfx1250) — compile-verified
//
#include <hip/hip_runtime.h>
#include <hip/hip_bf16.h>

typedef __attribute__((ext_vector_type(2))) float v2f;
typedef __attribute__((ext_vector_type(8))) float v8f;

#define ALIGN_HW 14
#define FM_H 64
#define FM_W 64
#define FM_C 256
#define ROIS_PER_BATCH 128

// One wave32 handles one (box, i) output row: 14 pixels x 256 channels.
// out[j, c] = sum_q w_q(j) * corner_q(j, c) computed as a chained-K sparse GEMM
// with V_WMMA_F32_16X16X4_F32:  D[16x16] += A_kk[16x4] * B_kk[4x16], kk = 0..13.
__global__ __launch_bounds__(256) void roialign_wmma_kernel(
    const float* __restrict__ fmap,   // [8, 64, 64, 256]
    const float* __restrict__ rois,   // [8, 128, 4] -> flat [1024, 4]
    float* __restrict__ out,          // [1024, 14, 14, 256]
    int nwaves)                       // 1024 * 14
{
    const int lane = threadIdx.x & 31;
    const int wave = blockIdx.x * (blockDim.x >> 5) + (threadIdx.x >> 5);
    if (wave >= nwaves) return;                      // wave-uniform branch

    const int box = wave / ALIGN_HW;
    const int i   = wave - box * ALIGN_HW;
    const int batch = box >> 7;                      // box_indices = box // 128

    // ROI (wave-uniform; compiler scalarizes)
    const float ry1 = rois[box * 4 + 0];
    const float rx1 = rois[box * 4 + 1];
    const float ry2 = rois[box * 4 + 2];
    const float rx2 = rois[box * 4 + 3];

    const float h_scale = (ry2 - ry1) * (float)(FM_H - 1) / (float)(ALIGN_HW - 1);
    const float w_scale = (rx2 - rx1) * (float)(FM_W - 1) / (float)(ALIGN_HW - 1);

    // y sampling (uniform per wave)
    const float in_y  = ry1 * (float)(FM_H - 1) + (float)i * h_scale;
    const bool  vy    = (in_y >= 0.0f) && (in_y <= (float)(FM_H - 1));
    const float fy    = floorf(in_y);
    const float ylerp = in_y - fy;
    const int   ytop  = (int)fminf(fmaxf(fy, 0.0f), (float)(FM_H - 1));
    const int   ybot  = (int)fminf(fmaxf(ceilf(in_y), 0.0f), (float)(FM_H - 1));

    // Per-lane role: lanes 0-15 carry the TOP feature row (A K=0/1, B K=0/1),
    // lanes 16-31 the BOTTOM row (A K=2/3, B K=2/3). myj = this lane's pixel.
    const bool lower = (lane < 16);
    const int  myj   = lane & 15;

    // x sampling for this lane's pixel j = myj (used only for the A weights)
    const float in_xm  = rx1 * (float)(FM_W - 1) + (float)myj * w_scale;
    const bool  vxm    = (in_xm >= 0.0f) && (in_xm <= (float)(FM_W - 1));
    const float xlerpm = in_xm - floorf(in_xm);

    const float wy   = lower ? (1.0f - ylerp) : ylerp;
    const float mask = ((vy && vxm) && (myj < ALIGN_HW)) ? 1.0f : 0.0f;
    const float a0   = wy * (1.0f - xlerpm) * mask;  // weight for left corner
    const float a1   = wy * xlerpm * mask;           // weight for right corner

    // Feature-row base for this lane: fmap[batch, ysel, :, myj + cbase]
    const int ysel = lower ? ytop : ybot;
    const float* rowbase =
        fmap + ((size_t)(batch * FM_H + ysel)) * (size_t)(FM_W * FM_C) + myj;

    // Output base: out[box, i, :, myj + cbase]
    float* outp = out + ((size_t)(box * ALIGN_HW + i)) * (size_t)(ALIGN_HW * FM_C) + myj;

    for (int t = 0; t < FM_C / 16; ++t) {
        const int cbase = t * 16;
        v8f acc = {0.f, 0.f, 0.f, 0.f, 0.f, 0.f, 0.f, 0.f};

        #pragma unroll
        for (int kk = 0; kk < ALIGN_HW; ++kk) {
            // x indices for pixel j = kk (uniform across lanes)
            const float in_x = rx1 * (float)(FM_W - 1) + (float)kk * w_scale;
            const int x0 = (int)fminf(fmaxf(floorf(in_x), 0.0f), (float)(FM_W - 1));
            const int x1 = (int)fminf(fmaxf(ceilf(in_x),  0.0f), (float)(FM_W - 1));

            // A slice: nonzero only for lanes whose pixel is kk
            const bool sel = (myj == kk);
            v2f a;
            a.x = sel ? a0 : 0.0f;
            a.y = sel ? a1 : 0.0f;

            // B slice: v0 = left corner (K=0 top / K=2 bottom),
            //          v1 = right corner (K=1 top / K=3 bottom), N = channel
            v2f b;
            b.x = rowbase[(size_t)x0 * FM_C + cbase];
            b.y = rowbase[(size_t)x1 * FM_C + cbase];

            acc = __builtin_amdgcn_wmma_f32_16x16x4_f32(
                /*neg_a=*/false, a, /*neg_b=*/false, b,
                /*c_mod=*/(short)0, acc, /*reuse_a=*/false, /*reuse_b=*/false);
        }

        // D layout: VGPR r -> row M = r (lanes 0-15) / r+8 (lanes 16-31), N = lane%16
        #pragma unroll
        for (int r = 0; r < 8; ++r) {
            const int j = r + (lower ? 0 : 8);
            if (j < ALIGN_HW) {
                outp[(size_t)j * FM_C + cbase] = acc[r];
            }
        }
    }
}

extern "C" void kernel_launch(void* const* d_in, const int* in_sizes, int n_in,
                              void* d_out, int out_size, void* d_ws, size_t ws_size,
                              hipStream_t stream) {
    const float* fmap = (const float*)d_in[0];   // [8,64,64,256] f32
    const float* rois = (const float*)d_in[1];   // [8,128,4]    f32
    float* out = (float*)d_out;                  // [1024,14,14,256] f32

    const int nboxes = in_sizes[1] / 4;          // 1024
    const int nwaves = nboxes * ALIGN_HW;        // 14336
    const int wavesPerBlock = 8;                 // 256 threads = 8 wave32
    const int blocks = (nwaves + wavesPerBlock - 1) / wavesPerBlock;

    hipLaunchKernelGGL(roialign_wmma_kernel, dim3(blocks), dim3(256), 0, stream,
                       fmap, rois, out, nwaves);
}